// MambaBlock_86320252715478
// MI455X (gfx1250) — compile-verified
//
#include <hip/hip_runtime.h>
#include <hip/hip_bf16.h>
#include <math.h>

typedef float v2f __attribute__((ext_vector_type(2)));
typedef float v8f __attribute__((ext_vector_type(8)));
typedef unsigned int u32x4 __attribute__((ext_vector_type(4)));
typedef int i32x4 __attribute__((ext_vector_type(4)));
typedef int i32x8 __attribute__((ext_vector_type(8)));

#define D_MODEL  1024
#define N_STATES 16
#define N_GROUPS 8
#define K_CONV   4
#define BATCH    2
#define LENGTH   2048
#define M_TOT    (BATCH * LENGTH)   // 4096 rows
#define BK       32                 // K-chunk staged in LDS
#define LDA_PAD  34                 // LDS row stride in floats (TDM pad: +2 DWORDs per 32)
#define MTILES   4
#define MROWS    (MTILES * 16)      // 64 M-rows per block

// ---------------------------------------------------------------------------
// TDM: issue TENSOR_LOAD_TO_LDS of a (MROWS x BK) fp32 tile of X starting at
// element (m0, k0), row stride K elements, into LDS byte offset lds_off with
// a 2-DWORD pad every 32 DWORDs (row stride LDA_PAD floats in LDS).
// Must be executed by exactly one wave (wave-uniform branch), EXEC all ones.
// ---------------------------------------------------------------------------
__device__ __forceinline__ void tdm_load_tile(const float* X, int ldK,
                                              int m0, int k0, unsigned lds_off) {
  unsigned long long ga =
      (unsigned long long)(size_t)(X + (size_t)m0 * ldK + k0);
  // D# group 0: count=1 | lds_addr | global_addr[56:0] | type=2
  u32x4 g0 = { 1u, lds_off, (unsigned)(ga & 0xFFFFFFFFu),
               (unsigned)((ga >> 32) & 0x01FFFFFFu) | (2u << 30) };
  // D# group 1:
  //  s0: data_size=2(4B)<<16 | pad_enable<<20 | pad_interval=4(32dw)<<22 | pad_amount=1(2dw)<<25
  //  s1: abar_addr=0 | tensor_dim0.lo16<<16      (tensor_dim0 = BK)
  //  s2: tensor_dim0.hi16=0 | tensor_dim1.lo16<<16 (tensor_dim1 = MROWS)
  //  s3: tensor_dim1.hi16=0 | tile_dim0<<16       (tile_dim0 = BK)
  //  s4: tile_dim1 | tile_dim2=0<<16              (tile_dim1 = MROWS)
  //  s5: tensor_dim0_stride.lo32 = ldK
  //  s6: tensor_dim0_stride.hi16=0 | tensor_dim1_stride.lo16=0
  i32x8 g1 = { (2 << 16) | (1 << 20) | (4 << 22) | (1 << 25),
               (BK & 0xFFFF) << 16,
               (MROWS & 0xFFFF) << 16,
               (BK & 0xFFFF) << 16,
               (MROWS & 0xFFFF),
               ldK, 0, 0 };
  i32x4 z4 = { 0, 0, 0, 0 };
#if defined(__clang_major__) && (__clang_major__ >= 23)
  i32x8 z8 = { 0, 0, 0, 0, 0, 0, 0, 0 };
  __builtin_amdgcn_tensor_load_to_lds(g0, g1, z4, z4, z8, 0);
#else
  __builtin_amdgcn_tensor_load_to_lds(g0, g1, z4, z4, 0);
#endif
}

// ---------------------------------------------------------------------------
// fp32 WMMA GEMM:  OUT = epilogue( X[M x K] @ W[N x K]^T (+ bias) )
// Block: 256 threads = 8 waves; block tile = 64(M) x 128(N).
// Each wave owns 4 stacked 16x16 output tiles (B fragment reused 4x).
// A tiles double-buffered in LDS, filled by the Tensor Data Mover.
// EPI 0: split 2048 cols -> u and sv = silu(v);  EPI 1: softplus;  EPI 2: +bias
// ---------------------------------------------------------------------------
template<int EPI>
__global__ __launch_bounds__(256) void wmma_gemm_kernel(
    const float* __restrict__ X, const float* __restrict__ W,
    const float* __restrict__ bias,
    float* __restrict__ out0, float* __restrict__ out1,
    int K, int ldOut)
{
  __shared__ float As[2][MROWS * LDA_PAD];

  const int tid  = threadIdx.x;
  const int lane = tid & 31;
  const int wave = tid >> 5;
  const int m0   = blockIdx.x * MROWS;
  const int n0   = blockIdx.y * 128 + wave * 16;
  const int nn   = n0 + (lane & 15);       // output column (B/C/D fragments)
  const int koff = (lane >> 4) << 1;       // 0 (lanes 0-15) or 2 (lanes 16-31)
  const int arow = lane & 15;              // M row within a 16x16 tile

  v8f acc[MTILES];
  #pragma unroll
  for (int t = 0; t < MTILES; ++t) acc[t] = (v8f){};

  const float* __restrict__ wrow = W + (size_t)nn * K;

  // prologue: DMA first A chunk into buffer 0
  if (wave == 0)
    tdm_load_tile(X, K, m0, 0, (unsigned)(size_t)(&As[0][0]));

  int buf = 0;
  for (int k0 = 0; k0 < K; k0 += BK, buf ^= 1) {
    if (wave == 0) {
      __builtin_amdgcn_s_wait_tensorcnt(0);          // current buffer landed
      if (k0 + BK < K)                               // overlap next DMA w/ math
        tdm_load_tile(X, K, m0, k0 + BK,
                      (unsigned)(size_t)(&As[buf ^ 1][0]));
    }
    __syncthreads();

    if (k0 + BK < K)                                 // keep W streaming warm
      __builtin_prefetch(&wrow[k0 + BK], 0, 3);

    #pragma unroll
    for (int kk = 0; kk < BK; kk += 4) {
      v2f b;
      b.x = wrow[k0 + kk + koff];
      b.y = wrow[k0 + kk + koff + 1];
      #pragma unroll
      for (int t = 0; t < MTILES; ++t) {
        v2f a;
        a.x = As[buf][(t * 16 + arow) * LDA_PAD + kk + koff];
        a.y = As[buf][(t * 16 + arow) * LDA_PAD + kk + koff + 1];
        acc[t] = __builtin_amdgcn_wmma_f32_16x16x4_f32(
            false, a, false, b, (short)0, acc[t], false, false);
      }
    }
    __syncthreads();
  }

  // Epilogue. C/D layout: VGPR r holds M = r (lanes 0-15) or r+8 (lanes 16-31).
  #pragma unroll
  for (int t = 0; t < MTILES; ++t) {
    #pragma unroll
    for (int r = 0; r < 8; ++r) {
      const int row = m0 + t * 16 + r + ((lane >> 4) << 3);
      float val = acc[t][r];
      if (EPI == 0) {
        val += bias[nn];
        if (nn < D_MODEL) {
          out0[(size_t)row * ldOut + nn] = val;                     // u
        } else {
          const float s = val / (1.0f + __expf(-val));              // silu(v)
          out1[(size_t)row * ldOut + (nn - D_MODEL)] = s;
        }
      } else if (EPI == 1) {
        const float sp = (val > 20.0f) ? val : log1pf(__expf(val)); // softplus
        out0[(size_t)row * ldOut + nn] = sp;
      } else {
        out0[(size_t)row * ldOut + nn] = val + bias[nn];            // Bmat
      }
    }
  }
}

// ---------------------------------------------------------------------------
// Depthwise causal conv (K=4) along L, + SiLU.  One thread per (m, d).
// ---------------------------------------------------------------------------
__global__ __launch_bounds__(256) void conv_silu_kernel(
    const float* __restrict__ u, const float* __restrict__ conv_w,
    const float* __restrict__ conv_b, float* __restrict__ uc)
{
  const int idx = blockIdx.x * 256 + threadIdx.x;   // over M_TOT * D
  const int d = idx & (D_MODEL - 1);
  const int m = idx >> 10;
  const int l = m & (LENGTH - 1);
  const int b = m >> 11;
  float acc = conv_b[d];
  #pragma unroll
  for (int j = 0; j < K_CONV; ++j) {
    const int ll = l - (K_CONV - 1) + j;
    if (ll >= 0)
      acc = fmaf(conv_w[d * K_CONV + j],
                 u[(size_t)(b * LENGTH + ll) * D_MODEL + d], acc);
  }
  uc[idx] = acc / (1.0f + __expf(-acc));            // silu
}

// ---------------------------------------------------------------------------
// Selective scan. 16 lanes = one (b,d) channel (one lane per state n),
// 2 channels per wave32. Time is processed in 16-step chunks: delta/u_conv/sv
// are loaded lane-parallel once per chunk and broadcast via shfl, so the
// serial recurrence only touches Bmat per step. Fuses y * silu(v).
// ---------------------------------------------------------------------------
__global__ __launch_bounds__(256) void scan_kernel(
    const float* __restrict__ delta, const float* __restrict__ a_log,
    const float* __restrict__ Bmat,  const float* __restrict__ c_param,
    const float* __restrict__ uc,    const float* __restrict__ sv,
    float* __restrict__ out)
{
  const int tid  = threadIdx.x;
  const int lane = tid & 31;
  const int wave = tid >> 5;
  const int n    = lane & 15;                        // state index / chunk slot
  const int ch   = lane >> 4;                        // channel within wave
  const int idx  = (blockIdx.x * 8 + wave) * 2 + ch; // flat (b,d)
  const int b    = idx >> 10;                        // / D_MODEL
  const int d    = idx & (D_MODEL - 1);
  const int g    = d >> 7;                           // group = d / 128

  const float An = -__expf(a_log[d * N_STATES + n]); // A = -exp(a_log)
  const float cn = c_param[d * N_STATES + n];

  float h = 0.0f;
  const int base = b * LENGTH;
  for (int l0 = 0; l0 < LENGTH; l0 += 16) {
    // lane-parallel chunk loads: lane n holds timestep l0+n of this channel
    const size_t mrow = (size_t)(base + l0 + n) * D_MODEL + d;
    const float dch = delta[mrow];
    const float uch = uc[mrow];
    const float svv = sv[mrow];
    if (l0 + 16 < LENGTH) {
      __builtin_prefetch(&delta[mrow + (size_t)16 * D_MODEL], 0, 3);
      __builtin_prefetch(&uc[mrow + (size_t)16 * D_MODEL], 0, 3);
      __builtin_prefetch(&Bmat[(size_t)(base + l0 + 16) * 128 + g * 16 + n], 0, 3);
    }
    float yout = 0.0f;
    #pragma unroll
    for (int t = 0; t < 16; ++t) {
      const float dlt = __shfl(dch, t, 16);          // broadcast step t
      const float u1  = __shfl(uch, t, 16);
      const float Bv  = Bmat[(size_t)(base + l0 + t) * 128 + g * 16 + n];
      h = fmaf(__expf(dlt * An), h, (dlt * u1) * Bv); // h = dA*h + dBu
      float y = h * cn;
      y += __shfl_xor(y, 8, 16);                     // reduce over 16 states
      y += __shfl_xor(y, 4, 16);
      y += __shfl_xor(y, 2, 16);
      y += __shfl_xor(y, 1, 16);
      if (n == t) yout = y * svv;                    // this lane's timestep
    }
    out[mrow] = yout;                                // lane-parallel store
  }
}

// ---------------------------------------------------------------------------
extern "C" void kernel_launch(void* const* d_in, const int* in_sizes, int n_in,
                              void* d_out, int out_size, void* d_ws, size_t ws_size,
                              hipStream_t stream) {
  const float* x            = (const float*)d_in[0];
  const float* in_proj_w    = (const float*)d_in[1];
  const float* in_proj_b    = (const float*)d_in[2];
  const float* delta_proj_w = (const float*)d_in[3];
  const float* a_log        = (const float*)d_in[4];
  const float* b_proj_w     = (const float*)d_in[5];
  const float* b_proj_b     = (const float*)d_in[6];
  const float* c_param      = (const float*)d_in[7];
  const float* conv_w       = (const float*)d_in[8];
  const float* conv_b       = (const float*)d_in[9];
  float* out = (float*)d_out;

  float* ws = (float*)d_ws;
  const size_t MD = (size_t)M_TOT * D_MODEL;   // 4M elements
  float* u     = ws;              // [M, D]
  float* sv    = ws + 1 * MD;     // silu(v)   [M, D]
  float* delta = ws + 2 * MD;     // [M, D]
  float* uconv = ws + 3 * MD;     // [M, D]
  float* Bm    = ws + 4 * MD;     // [M, 128]

  dim3 blk(256);

  // 1) in_proj: [4096 x 2048] = x @ in_proj_w^T; split into u and silu(v)
  wmma_gemm_kernel<0><<<dim3(M_TOT / MROWS, 2048 / 128), blk, 0, stream>>>(
      x, in_proj_w, in_proj_b, u, sv, D_MODEL, D_MODEL);

  // 2) delta = softplus(u @ delta_proj_w^T): [4096 x 1024]
  wmma_gemm_kernel<1><<<dim3(M_TOT / MROWS, 1024 / 128), blk, 0, stream>>>(
      u, delta_proj_w, nullptr, delta, nullptr, D_MODEL, D_MODEL);

  // 3) Bmat = u @ b_proj_w^T + b: [4096 x 128]
  wmma_gemm_kernel<2><<<dim3(M_TOT / MROWS, 1), blk, 0, stream>>>(
      u, b_proj_w, b_proj_b, Bm, nullptr, D_MODEL, 128);

  // 4) u_conv = silu(causal depthwise conv(u))
  conv_silu_kernel<<<dim3((unsigned)(MD / 256)), blk, 0, stream>>>(
      u, conv_w, conv_b, uconv);

  // 5) selective scan + output gating (2048 channels, 16 per block)
  scan_kernel<<<dim3((BATCH * D_MODEL) / 16), blk, 0, stream>>>(
      delta, a_log, Bm, c_param, uconv, sv, out);
}